// GIN_55095840473501
// MI455X (gfx1250) — compile-verified
//
#include <hip/hip_runtime.h>
#include <hip/hip_bf16.h>
#include <math.h>

// ---------------------------------------------------------------------------
// GIN forward on MI455X (gfx1250, wave32).
// Dense math uses V_WMMA_F32_16X16X4_F32 (fp32 matrix core path).
// ---------------------------------------------------------------------------

#define N_NODES 100000
#define E_EDGES 3200000
#define F_INF   128
#define HDIM    32
#define BGR     128
#define NCLS    8
#define BN_EPS  1e-5f

typedef float v2f __attribute__((ext_vector_type(2)));
typedef float v8f __attribute__((ext_vector_type(8)));

__device__ __forceinline__ v8f wmma_k4(v2f a, v2f b, v8f c) {
  // D = A(16x4,f32) * B(4x16,f32) + C(16x16,f32)
  return __builtin_amdgcn_wmma_f32_16x16x4_f32(false, a, false, b, (short)0, c,
                                               false, false);
}

// -------------------------------- utils ------------------------------------
__global__ void fill_zero_kernel(float* __restrict__ p, int n) {
  int i = blockIdx.x * blockDim.x + threadIdx.x;
  if (i < n) p[i] = 0.0f;
}

// ---------------------- layer-0 projection: xw = x @ w1_0 -------------------
// grid: ceil(N/64) blocks x 128 threads (4 waves, 16 rows/wave, 32 cols).
#define XSTR 132  // padded LDS row stride (floats): bank-conflict free, 16B aligned rows
__global__ __launch_bounds__(128) void gemm_xw_kernel(
    const float* __restrict__ x, const float* __restrict__ w,
    float* __restrict__ xw) {
  __shared__ float wt[HDIM * XSTR];      // W^T: wt[col*XSTR + k], k in [0,128)
  __shared__ float xs[4][16 * XSTR];     // per-wave 16x128 X tile

  const int tid = threadIdx.x, wave = tid >> 5, lane = tid & 31;

  for (int idx = tid; idx < F_INF * HDIM; idx += 128) {
    int k = idx >> 5, col = idx & 31;
    wt[col * XSTR + k] = w[idx];
  }

  const int row0 = blockIdx.x * 64 + wave * 16;
  for (int r = 0; r < 16; ++r) {
    int row = row0 + r;
    if (row >= N_NODES) row = N_NODES - 1;  // clamp; tail rows never stored
    const float4 v = *(const float4*)&x[(size_t)row * F_INF + lane * 4];
    *(float4*)&xs[wave][r * XSTR + lane * 4] = v;
  }
  __syncthreads();

  const int m = lane & 15;
  const int kh = (lane >> 4) * 2;   // K-half offset per WMMA A/B layout
  const int roff = (lane >> 4) * 8; // C-layout row offset

  v8f acc0 = {0, 0, 0, 0, 0, 0, 0, 0};
  v8f acc1 = {0, 0, 0, 0, 0, 0, 0, 0};
  const float* az = &xs[wave][m * XSTR + kh];
  const float* b0 = &wt[m * XSTR + kh];          // cols 0..15
  const float* b1 = &wt[(m + 16) * XSTR + kh];   // cols 16..31
#pragma unroll
  for (int k = 0; k < F_INF; k += 4) {
    v2f a  = *(const v2f*)(az + k);
    v2f bb0 = *(const v2f*)(b0 + k);
    v2f bb1 = *(const v2f*)(b1 + k);
    acc0 = wmma_k4(a, bb0, acc0);
    acc1 = wmma_k4(a, bb1, acc1);
  }
#pragma unroll
  for (int r = 0; r < 8; ++r) {
    int row = row0 + r + roff;
    if (row < N_NODES) {
      xw[(size_t)row * HDIM + m]      = acc0[r];
      xw[(size_t)row * HDIM + m + 16] = acc1[r];
    }
  }
}

// ----------------------------- edge aggregation -----------------------------
// one wave per edge: lane = feature. Coalesced 128B gather + 128B atomic burst.
__global__ void agg_kernel(const int* __restrict__ src, const int* __restrict__ dst,
                           const float* __restrict__ feat, float* __restrict__ agg) {
  unsigned idx = blockIdx.x * blockDim.x + threadIdx.x;
  unsigned e = idx >> 5;
  int f = idx & 31;
  if (e < E_EDGES) {
    if ((e + 2048u) < E_EDGES && f == 0) {
      __builtin_prefetch(&src[e + 2048u], 0, 0);
      __builtin_prefetch(&dst[e + 2048u], 0, 0);
    }
    int s = src[e], d = dst[e];
    float v = feat[(size_t)s * HDIM + f];
    atomicAdd(&agg[(size_t)d * HDIM + f], v);
  }
}

// -------------------- fused GIN MLP + ReLU + BN statistics ------------------
// firstGemm=1: a = relu(relu((base+agg)@w1 + b1) @ w2 + b2)
// firstGemm=0: a = relu(relu(base+agg+b1)        @ w2 + b2)   (layer 0 path)
#define ZSTR 36  // padded LDS stride for 16x32 tiles
__global__ __launch_bounds__(128) void gin_mlp_kernel(
    const float* __restrict__ base, const float* __restrict__ aggv,
    const float* __restrict__ w1, const float* __restrict__ b1,
    const float* __restrict__ w2, const float* __restrict__ b2,
    float* __restrict__ abuf, float* __restrict__ stats, int firstGemm) {
  __shared__ float wt1[HDIM * ZSTR];
  __shared__ float wt2[HDIM * ZSTR];
  __shared__ float bias1[HDIM], bias2[HDIM];
  __shared__ float zs[4][16 * ZSTR];
  __shared__ float ssum[64];

  const int tid = threadIdx.x, wave = tid >> 5, lane = tid & 31;

  for (int idx = tid; idx < HDIM * HDIM; idx += 128) {
    int k = idx >> 5, col = idx & 31;
    wt1[col * ZSTR + k] = w1[idx];
    wt2[col * ZSTR + k] = w2[idx];
  }
  if (tid < HDIM) { bias1[tid] = b1[tid]; bias2[tid] = b2[tid]; }
  if (tid < 64) ssum[tid] = 0.0f;
  __syncthreads();

  const int row0 = blockIdx.x * 64 + wave * 16;
  for (int r = 0; r < 16; ++r) {
    int row = row0 + r;
    if (row >= N_NODES) row = N_NODES - 1;
    float v = base[(size_t)row * HDIM + lane] + aggv[(size_t)row * HDIM + lane];
    if (!firstGemm) v = fmaxf(v + bias1[lane], 0.0f);
    zs[wave][r * ZSTR + lane] = v;
  }
  __syncthreads();

  const int m = lane & 15;
  const int kh = (lane >> 4) * 2;
  const int roff = (lane >> 4) * 8;
  const v8f zero8 = {0, 0, 0, 0, 0, 0, 0, 0};

  if (firstGemm) {
    v8f u0 = zero8, u1 = zero8;
#pragma unroll
    for (int k = 0; k < HDIM; k += 4) {
      v2f a  = *(const v2f*)&zs[wave][m * ZSTR + kh + k];
      v2f bb0 = *(const v2f*)&wt1[m * ZSTR + kh + k];
      v2f bb1 = *(const v2f*)&wt1[(m + 16) * ZSTR + kh + k];
      u0 = wmma_k4(a, bb0, u0);
      u1 = wmma_k4(a, bb1, u1);
    }
    float bv0 = bias1[m], bv1 = bias1[m + 16];
    __syncthreads();
#pragma unroll
    for (int r = 0; r < 8; ++r) {
      int rr = r + roff;
      zs[wave][rr * ZSTR + m]      = fmaxf(u0[r] + bv0, 0.0f);
      zs[wave][rr * ZSTR + m + 16] = fmaxf(u1[r] + bv1, 0.0f);
    }
    __syncthreads();
  }

  v8f acc0 = zero8, acc1 = zero8;
#pragma unroll
  for (int k = 0; k < HDIM; k += 4) {
    v2f a  = *(const v2f*)&zs[wave][m * ZSTR + kh + k];
    v2f bb0 = *(const v2f*)&wt2[m * ZSTR + kh + k];
    v2f bb1 = *(const v2f*)&wt2[(m + 16) * ZSTR + kh + k];
    acc0 = wmma_k4(a, bb0, acc0);
    acc1 = wmma_k4(a, bb1, acc1);
  }

  const float bv0 = bias2[m], bv1 = bias2[m + 16];
  float s0 = 0, q0 = 0, s1 = 0, q1 = 0;
#pragma unroll
  for (int r = 0; r < 8; ++r) {
    int row = row0 + r + roff;
    float v0 = fmaxf(acc0[r] + bv0, 0.0f);
    float v1 = fmaxf(acc1[r] + bv1, 0.0f);
    if (row < N_NODES) {  // mask tail rows out of BN stats
      abuf[(size_t)row * HDIM + m]      = v0;
      abuf[(size_t)row * HDIM + m + 16] = v1;
      s0 += v0; q0 += v0 * v0;
      s1 += v1; q1 += v1 * v1;
    }
  }
  atomicAdd(&ssum[m], s0);       atomicAdd(&ssum[32 + m], q0);
  atomicAdd(&ssum[m + 16], s1);  atomicAdd(&ssum[32 + m + 16], q1);
  __syncthreads();
  if (tid < 64) atomicAdd(&stats[tid], ssum[tid]);
}

// ------------------------------ batch norm ----------------------------------
__global__ void bn_finalize_kernel(float* __restrict__ stats) {
  int f = threadIdx.x;  // 32 threads
  float mean = stats[f] * (1.0f / (float)N_NODES);
  float var  = stats[32 + f] * (1.0f / (float)N_NODES) - mean * mean;
  stats[64 + f] = mean;
  stats[96 + f] = rsqrtf(var + BN_EPS);
}

__global__ void bn_apply_kernel(const float* __restrict__ a,
                                const float* __restrict__ stats,
                                const float* __restrict__ g,
                                const float* __restrict__ be,
                                float* __restrict__ h) {
  int idx = blockIdx.x * blockDim.x + threadIdx.x;
  if (idx < N_NODES * HDIM) {
    int f = idx & 31;
    h[idx] = g[f] * (a[idx] - stats[64 + f]) * stats[96 + f] + be[f];
  }
}

// ------------------------------ mean pool -----------------------------------
__global__ void pool_kernel(const float* __restrict__ h, const int* __restrict__ batch,
                            float* __restrict__ pooled, float* __restrict__ cnt) {
  int idx = blockIdx.x * blockDim.x + threadIdx.x;
  if (idx < N_NODES * HDIM) {
    int i = idx >> 5, f = idx & 31;
    int b = batch[i];
    atomicAdd(&pooled[b * HDIM + f], h[idx]);
    if (f == 0) atomicAdd(&cnt[b], 1.0f);
  }
}

// ------------------------------ classifier head -----------------------------
__global__ void head_kernel(const float* __restrict__ pooled, const float* __restrict__ cnt,
                            const float* __restrict__ fc1w, const float* __restrict__ fc1b,
                            const float* __restrict__ fc2w, const float* __restrict__ fc2b,
                            float* __restrict__ out) {
  int g = threadIdx.x;  // 128 threads, one graph each
  if (g >= BGR) return;
  float inv = 1.0f / fmaxf(cnt[g], 1.0f);
  float p[HDIM];
  for (int k = 0; k < HDIM; ++k) p[k] = pooled[g * HDIM + k] * inv;
  float u[HDIM];
  for (int j = 0; j < HDIM; ++j) {
    float s = fc1b[j];
    for (int k = 0; k < HDIM; ++k) s += p[k] * fc1w[k * HDIM + j];
    u[j] = fmaxf(s, 0.0f);
  }
  float o[NCLS];
  float mx = -INFINITY;
  for (int j = 0; j < NCLS; ++j) {
    float s = fc2b[j];
    for (int k = 0; k < HDIM; ++k) s += u[k] * fc2w[k * NCLS + j];
    o[j] = s;
    mx = fmaxf(mx, s);
  }
  float se = 0.0f;
  for (int j = 0; j < NCLS; ++j) se += expf(o[j] - mx);
  float lse = mx + logf(se);
  for (int j = 0; j < NCLS; ++j) out[g * NCLS + j] = o[j] - lse;
}

// ---------------------------------------------------------------------------
extern "C" void kernel_launch(void* const* d_in, const int* in_sizes, int n_in,
                              void* d_out, int out_size, void* d_ws, size_t ws_size,
                              hipStream_t stream) {
  (void)in_sizes; (void)n_in; (void)out_size; (void)ws_size;

  const float* x     = (const float*)d_in[0];
  const int*   ei    = (const int*)d_in[1];
  const int*   src   = ei;
  const int*   dst   = ei + E_EDGES;
  const int*   batch = (const int*)d_in[2];
  const float* w1_0  = (const float*)d_in[3];
  const float* b1_0  = (const float*)d_in[4];
  const float* w2_0  = (const float*)d_in[5];
  const float* b2_0  = (const float*)d_in[6];
  const float* g_0   = (const float*)d_in[7];
  const float* be_0  = (const float*)d_in[8];
  const float* w1s   = (const float*)d_in[9];
  const float* b1s   = (const float*)d_in[10];
  const float* w2s   = (const float*)d_in[11];
  const float* b2s   = (const float*)d_in[12];
  const float* gs    = (const float*)d_in[13];
  const float* bes   = (const float*)d_in[14];
  const float* fc1w  = (const float*)d_in[15];
  const float* fc1b  = (const float*)d_in[16];
  const float* fc2w  = (const float*)d_in[17];
  const float* fc2b  = (const float*)d_in[18];
  float* out = (float*)d_out;

  const int NH = N_NODES * HDIM;  // 3.2M floats
  float* ws     = (float*)d_ws;   // needs ~51.3 MB of scratch
  float* xw     = ws;             // N x 32   (layer-0 projection x@W1)
  float* agg    = xw + NH;        // N x 32
  float* abuf   = agg + NH;       // N x 32   (pre-BN activations)
  float* hbuf   = abuf + NH;      // N x 32   (post-BN activations)
  float* stats  = hbuf + NH;      // [sum32 | sumsq32 | mean32 | invstd32]
  float* pooled = stats + 128;    // B x 32
  float* cnt    = pooled + BGR * HDIM;  // B  (contiguous with pooled)

  const int gemmBlocks = (N_NODES + 63) / 64;
  const int elemBlocks = (NH + 255) / 256;
  const int aggBlocks  = (E_EDGES * 32) / 256;

  // ---- layer 0: project x to 32-dim, then aggregate in 32-dim space -------
  fill_zero_kernel<<<elemBlocks, 256, 0, stream>>>(agg, NH);
  fill_zero_kernel<<<1, 64, 0, stream>>>(stats, 64);
  gemm_xw_kernel<<<gemmBlocks, 128, 0, stream>>>(x, w1_0, xw);
  agg_kernel<<<aggBlocks, 256, 0, stream>>>(src, dst, xw, agg);
  gin_mlp_kernel<<<gemmBlocks, 128, 0, stream>>>(xw, agg, w1_0, b1_0, w2_0, b2_0,
                                                 abuf, stats, 0);
  bn_finalize_kernel<<<1, 32, 0, stream>>>(stats);
  bn_apply_kernel<<<elemBlocks, 256, 0, stream>>>(abuf, stats, g_0, be_0, hbuf);

  // ---- layers 1..3 --------------------------------------------------------
  for (int i = 0; i < 3; ++i) {
    fill_zero_kernel<<<elemBlocks, 256, 0, stream>>>(agg, NH);
    fill_zero_kernel<<<1, 64, 0, stream>>>(stats, 64);
    agg_kernel<<<aggBlocks, 256, 0, stream>>>(src, dst, hbuf, agg);
    gin_mlp_kernel<<<gemmBlocks, 128, 0, stream>>>(
        hbuf, agg, w1s + i * HDIM * HDIM, b1s + i * HDIM,
        w2s + i * HDIM * HDIM, b2s + i * HDIM, abuf, stats, 1);
    bn_finalize_kernel<<<1, 32, 0, stream>>>(stats);
    bn_apply_kernel<<<elemBlocks, 256, 0, stream>>>(abuf, stats, gs + i * HDIM,
                                                    bes + i * HDIM, hbuf);
  }

  // ---- pool + head --------------------------------------------------------
  fill_zero_kernel<<<(BGR * HDIM + BGR + 255) / 256, 256, 0, stream>>>(pooled,
                                                                       BGR * HDIM + BGR);
  pool_kernel<<<elemBlocks, 256, 0, stream>>>(hbuf, batch, pooled, cnt);
  head_kernel<<<1, 128, 0, stream>>>(pooled, cnt, fc1w, fc1b, fc2w, fc2b, out);
}